// SE_32461362823188
// MI455X (gfx1250) — compile-verified
//
#include <hip/hip_runtime.h>

typedef __attribute__((ext_vector_type(2))) float v2f;
typedef __attribute__((ext_vector_type(4))) float v4f;
typedef __attribute__((ext_vector_type(8))) float v8f;

#define B_  32
#define C_  256
#define MID_ 64
#define HW_ 3136          // 56*56
#define HW4_ 784          // HW/4
#define PLANES_ (B_*C_)   // 8192

// ---------------------------------------------------------------------------
// Pass 1: global average pool. One wave (32 lanes) per (b,c) plane.
// 3136 contiguous floats -> 784 float4; 24 full strided iters + 16-lane tail.
// ---------------------------------------------------------------------------
__global__ void se_pool_kernel(const float* __restrict__ x,
                               float* __restrict__ pooled) {
    const int wave  = threadIdx.x >> 5;
    const int lane  = threadIdx.x & 31;
    const int plane = blockIdx.x * 8 + wave;   // 1024 blocks * 8 waves = 8192

    const v4f* xv = (const v4f*)(x + (size_t)plane * HW_);
    float s = 0.0f;
#pragma unroll
    for (int i = 0; i < 24; ++i) {             // 24*32 = 768 vec4
        v4f v = xv[lane + i * 32];
        s += v.x + v.y + v.z + v.w;
    }
    if (lane < 16) {                           // tail: vec4 768..783
        v4f v = xv[768 + lane];
        s += v.x + v.y + v.z + v.w;
    }
#pragma unroll
    for (int off = 16; off >= 1; off >>= 1)
        s += __shfl_xor(s, off, 32);
    if (lane == 0)
        pooled[plane] = s * (1.0f / (float)HW_);
}

// ---------------------------------------------------------------------------
// Pass 2: SE MLP via V_WMMA_F32_16X16X4_F32 (fp32, full precision).
// Single workgroup, 256 threads = 8 waves.
//   GEMM1: pooled[32,256] x w1^T[256,64] -> h[32,64]   (8 tiles, 1/wave)
//   GEMM2: h[32,64]       x w2^T[64,256] -> y[32,256]  (32 tiles, 4/wave)
// A 16x4 f32 layout: lane m=lane&15, kh=lane>>4; VGPR0/1 = K = 2*kh, 2*kh+1.
// B 4x16 f32 layout mirrors A with n=lane&15.
// D 16x16: VGPR r -> row r + 8*(lane>>4), col lane&15.
// ---------------------------------------------------------------------------
__global__ void se_mlp_kernel(const float* __restrict__ pooled,
                              const float* __restrict__ w1,   // [64,256]
                              const float* __restrict__ b1,   // [64]
                              const float* __restrict__ w2,   // [256,64]
                              const float* __restrict__ b2,   // [256]
                              float* __restrict__ gate) {     // [32,256]
    __shared__ float h_lds[B_ * MID_];   // 32*64 = 8 KB

    const int wave = threadIdx.x >> 5;
    const int lane = threadIdx.x & 31;
    const int m    = lane & 15;
    const int kg   = lane >> 4;          // 0: K+0/K+1 half, 1: K+2/K+3 half
    const int rowoff = kg ? 8 : 0;       // D-matrix row offset for this half

    // ---- GEMM1: tile (mt, nt), one per wave ----
    {
        const int mt = wave >> 2;        // 0..1
        const int nt = wave & 3;         // 0..3
        const int m0 = mt * 16;
        const int n0 = nt * 16;

        v8f acc = {};
        for (int k0 = 0; k0 < C_; k0 += 4) {
            const int k = k0 + 2 * kg;
            v2f a, b;
            a.x = pooled[(m0 + m) * C_ + k];
            a.y = pooled[(m0 + m) * C_ + k + 1];
            // B[k][n] = w1[n][k], w1 row-major [MID, C]
            b.x = w1[(n0 + m) * C_ + k];
            b.y = w1[(n0 + m) * C_ + k + 1];
            acc = __builtin_amdgcn_wmma_f32_16x16x4_f32(
                false, a, false, b, (short)0, acc, false, false);
        }
#pragma unroll
        for (int r = 0; r < 8; ++r) {
            const int row = m0 + r + rowoff;
            const int col = n0 + m;
            float v = acc[r] + b1[col];
            h_lds[row * MID_ + col] = v > 0.0f ? v : 0.0f;   // ReLU
        }
    }
    __syncthreads();

    // ---- GEMM2: 4 tiles per wave ----
    for (int t = wave; t < 32; t += 8) {
        const int mt = t >> 4;           // 0..1
        const int nt = t & 15;           // 0..15
        const int m0 = mt * 16;
        const int n0 = nt * 16;

        v8f acc = {};
#pragma unroll
        for (int k0 = 0; k0 < MID_; k0 += 4) {
            const int k = k0 + 2 * kg;
            v2f a, b;
            a.x = h_lds[(m0 + m) * MID_ + k];
            a.y = h_lds[(m0 + m) * MID_ + k + 1];
            // B[k][n] = w2[n][k], w2 row-major [C, MID]
            b.x = w2[(n0 + m) * MID_ + k];
            b.y = w2[(n0 + m) * MID_ + k + 1];
            acc = __builtin_amdgcn_wmma_f32_16x16x4_f32(
                false, a, false, b, (short)0, acc, false, false);
        }
#pragma unroll
        for (int r = 0; r < 8; ++r) {
            const int row = m0 + r + rowoff;
            const int col = n0 + m;
            float y = acc[r] + b2[col] + 3.0f;
            y = y < 0.0f ? 0.0f : (y > 6.0f ? 6.0f : y);     // relu6(y+3)
            gate[row * C_ + col] = y * 0.16666667f;          // hard-sigmoid
        }
    }
}

// ---------------------------------------------------------------------------
// Pass 3: out = x * gate[b,c]. One block per plane; gate load is
// block-uniform (scalar path). Non-temporal stores: output is never re-read.
// Second read of x should hit the 192 MB L2 (x is ~103 MB, loaded in pass 1).
// ---------------------------------------------------------------------------
__global__ void se_scale_kernel(const float* __restrict__ x,
                                const float* __restrict__ gate,
                                float* __restrict__ out) {
    const int plane = blockIdx.x;                // 8192 planes
    const float g = gate[plane];
    const v4f* xv = (const v4f*)(x + (size_t)plane * HW_);
    v4f* ov       = (v4f*)(out + (size_t)plane * HW_);
    for (int i = threadIdx.x; i < HW4_; i += blockDim.x) {
        v4f v = xv[i];
        v.x *= g; v.y *= g; v.z *= g; v.w *= g;
        __builtin_nontemporal_store(v, &ov[i]);
    }
}

extern "C" void kernel_launch(void* const* d_in, const int* in_sizes, int n_in,
                              void* d_out, int out_size, void* d_ws, size_t ws_size,
                              hipStream_t stream) {
    const float* x  = (const float*)d_in[0];   // [32,256,56,56]
    const float* w1 = (const float*)d_in[1];   // [64,256]
    const float* b1 = (const float*)d_in[2];   // [64]
    const float* w2 = (const float*)d_in[3];   // [256,64]
    const float* b2 = (const float*)d_in[4];   // [256]
    float* out = (float*)d_out;

    float* pooled = (float*)d_ws;              // 8192 floats
    float* gate   = pooled + PLANES_;          // 8192 floats (64 KB total ws)

    se_pool_kernel <<<PLANES_ / 8, 256, 0, stream>>>(x, pooled);
    se_mlp_kernel  <<<1,          256, 0, stream>>>(pooled, w1, b1, w2, b2, gate);
    se_scale_kernel<<<PLANES_,    256, 0, stream>>>(x, gate, out);
}